// VectorQuantizer_65661460022052
// MI455X (gfx1250) — compile-verified
//
#include <hip/hip_runtime.h>

// Problem constants (from reference): B=32, C=256, H=W=32, K=4096
#define HWSZ   1024          // H*W
#define CDIM   256
#define KCODES 4096
#define NROWS  32768         // B*H*W
#define CB_STRIDE 260        // LDS row stride (floats): 16 code rows hit distinct banks

typedef __attribute__((ext_vector_type(2))) float v2f;
typedef __attribute__((ext_vector_type(8))) float v8f;

// ---------------------------------------------------------------------------
// Kernel 1: codebook squared norms  ||e_k||^2  -> ws
// ---------------------------------------------------------------------------
__global__ void vq_norms_kernel(const float* __restrict__ cb,
                                float* __restrict__ norms) {
    int k = blockIdx.x * blockDim.x + threadIdx.x;   // 0..4095
    const float4* row = (const float4*)(cb + (size_t)k * CDIM);
    float s = 0.f;
#pragma unroll 8
    for (int i = 0; i < CDIM / 4; ++i) {
        float4 v = row[i];
        s += v.x * v.x + v.y * v.y + v.z * v.z + v.w * v.w;
    }
    norms[k] = s;
}

// ---------------------------------------------------------------------------
// Kernel 2: WMMA distance GEMM + running argmin.
// Block = 256 threads = 8 waves; each wave owns 16 rows of xf (A in VGPRs).
// Codebook streamed via LDS, 16 codes per tile. argmin_k(||e||^2 - 2 x.e).
// ---------------------------------------------------------------------------
__global__ __launch_bounds__(256) void vq_argmin_kernel(
        const float* __restrict__ x,      // (B, C, H, W)
        const float* __restrict__ cb,     // (K, C)
        const float* __restrict__ norms,  // (K)
        float* __restrict__ ids_f,        // (N) ids as float (output 0)
        int*   __restrict__ ids_i) {      // (N) ids as int (workspace)
    __shared__ float cb_lds[16 * CB_STRIDE];

    const int tid  = threadIdx.x;
    const int lane = tid & 31;
    const int wave = tid >> 5;
    const int col  = lane & 15;          // column / A-row index within tile
    const int grp  = lane >> 4;          // 0 or 1
    const int kb   = grp << 1;           // K-subslot base for f32 16x16x4 frags

    // This lane's A-matrix row (row-major xf: n = b*HW + hw, element c at
    // x[(b*C + c)*HW + hw]).
    const int nrow = blockIdx.x * 128 + wave * 16 + col;
    const int bb = nrow >> 10;
    const int hw = nrow & (HWSZ - 1);

    // Preload all A fragments for this wave's 16x256 slice into VGPRs.
    // a[j] covers C-chunk [4j, 4j+4): this lane holds (4j+kb, 4j+kb+1).
    v2f a[64];
    const float* xb = x + ((size_t)bb * CDIM + kb) * HWSZ + hw;
#pragma unroll
    for (int j = 0; j < 64; ++j) {
        a[j].x = xb[(size_t)(4 * j) * HWSZ];
        a[j].y = xb[(size_t)(4 * j) * HWSZ + HWSZ];
    }

    float bestv[8];
    int   besti[8];
#pragma unroll
    for (int r = 0; r < 8; ++r) { bestv[r] = 3.4e38f; besti[r] = 0; }

    for (int t = 0; t < KCODES / 16; ++t) {
        __syncthreads();
        // Stage 16 codebook rows (contiguous 16*256 floats) into padded LDS.
        {
            const float4* src = (const float4*)(cb + (size_t)t * 16 * CDIM);
#pragma unroll
            for (int i = 0; i < 4; ++i) {
                int f4   = tid + i * 256;     // float4 index within tile
                int code = f4 >> 6;           // 64 float4 per code row
                int c4   = f4 & 63;
                *(float4*)&cb_lds[code * CB_STRIDE + c4 * 4] = src[f4];
            }
            if (t + 1 < KCODES / 16)   // global_prefetch_b8 of next tile
                __builtin_prefetch(cb + (size_t)(t + 1) * 16 * CDIM + tid * 16, 0, 0);
        }
        __syncthreads();

        // 16x16x256 f32 GEMM tile: 64 x v_wmma_f32_16x16x4_f32
        v8f acc = {0.f, 0.f, 0.f, 0.f, 0.f, 0.f, 0.f, 0.f};
        const float* bp = &cb_lds[col * CB_STRIDE + kb];
#pragma unroll
        for (int j = 0; j < 64; ++j) {
            v2f bf = *(const v2f*)(bp + 4 * j);
            acc = __builtin_amdgcn_wmma_f32_16x16x4_f32(
                false, a[j], false, bf, (short)0, acc, false, false);
        }

        const int   cid = t * 16 + col;
        const float nrm = norms[cid];
#pragma unroll
        for (int r = 0; r < 8; ++r) {
            float score = nrm - 2.0f * acc[r];
            bool better = score < bestv[r];
            bestv[r] = better ? score : bestv[r];
            besti[r] = better ? cid   : besti[r];
        }
    }

    // C/D layout: lanes 0-15 VGPR r -> row r (cols 0-15); lanes 16-31 -> row r+8.
    // Min-reduce each row across its 16 lanes (first-index tie-break = argmin).
#pragma unroll
    for (int r = 0; r < 8; ++r) {
        float v  = bestv[r];
        int   ix = besti[r];
#pragma unroll
        for (int m = 1; m < 16; m <<= 1) {
            float ov = __shfl_xor(v, m, 16);
            int   oi = __shfl_xor(ix, m, 16);
            if (ov < v || (ov == v && oi < ix)) { v = ov; ix = oi; }
        }
        if (col == 0) {
            int row = blockIdx.x * 128 + wave * 16 + r + grp * 8;
            ids_f[row] = (float)ix;
            ids_i[row] = ix;
        }
    }
}

// ---------------------------------------------------------------------------
// Kernel 3: gather emb[b][c][h][w] = code_book[ids[n]][c], float4 writes.
// ---------------------------------------------------------------------------
__global__ void vq_gather_kernel(const float* __restrict__ cb,
                                 const int* __restrict__ ids,
                                 float* __restrict__ emb) {
    int idx4 = blockIdx.x * blockDim.x + threadIdx.x;  // 0 .. B*C*HW/4 - 1
    int hw4 = idx4 & 255;            // HW/4 = 256
    int c   = (idx4 >> 8) & 255;
    int b   = idx4 >> 16;
    int n   = b * HWSZ + hw4 * 4;
    float4 o;
    o.x = cb[(size_t)ids[n + 0] * CDIM + c];
    o.y = cb[(size_t)ids[n + 1] * CDIM + c];
    o.z = cb[(size_t)ids[n + 2] * CDIM + c];
    o.w = cb[(size_t)ids[n + 3] * CDIM + c];
    *(float4*)&emb[((size_t)(b * CDIM + c)) * HWSZ + hw4 * 4] = o;
}

// ---------------------------------------------------------------------------
extern "C" void kernel_launch(void* const* d_in, const int* in_sizes, int n_in,
                              void* d_out, int out_size, void* d_ws, size_t ws_size,
                              hipStream_t stream) {
    const float* x  = (const float*)d_in[0];   // (32,256,32,32) f32
    const float* cb = (const float*)d_in[1];   // (4096,256) f32

    float* out   = (float*)d_out;              // [0,32768): ids; then emb
    float* norms = (float*)d_ws;                               // 16 KB
    int*   ids_i = (int*)((char*)d_ws + (size_t)KCODES * 4);   // 128 KB

    vq_norms_kernel <<<KCODES / 256, 256, 0, stream>>>(cb, norms);
    vq_argmin_kernel<<<NROWS / 128, 256, 0, stream>>>(x, cb, norms, out, ids_i);
    vq_gather_kernel<<<(32 * CDIM * HWSZ / 4) / 256, 256, 0, stream>>>(
        cb, ids_i, out + NROWS);
}